// GCNmyConv_10591389352439
// MI455X (gfx1250) — compile-verified
//
#include <hip/hip_runtime.h>

typedef float v2f __attribute__((ext_vector_type(2)));
typedef float v8f __attribute__((ext_vector_type(8)));

#define DFEAT 64

// ---------------------------------------------------------------- utilities
__global__ void fill_f32(float* __restrict__ p, float v, int n) {
    int i = blockIdx.x * blockDim.x + threadIdx.x;
    if (i < n) p[i] = v;
}

// ------------------------------------------------- h = x @ W^T + b  (WMMA)
// One wave per 16x16 output tile; K=64 -> 16 x v_wmma_f32_16x16x4_f32.
// A(16x4): lane r=lane&15 -> row m0+r, K = k + 2*(lane>=16) + {0,1}
// B(4x16): lane r        -> col n0+r (W row), same K split
// C/D: VGPR v -> (M = m0 + v + 8*(lane>=16), N = n0 + r)
__global__ void linear_wmma(const float* __restrict__ x,
                            const float* __restrict__ W,
                            const float* __restrict__ bias,
                            float* __restrict__ h,
                            int n_row_tiles) {
    const int wave = threadIdx.x >> 5;                 // 4 waves per block
    const int mt   = blockIdx.x * 4 + wave;            // row tile
    if (mt >= n_row_tiles) return;                     // wave-uniform guard
    const int nt   = blockIdx.y;                       // feature tile (0..3)
    const int m0   = mt * 16;
    const int n0   = nt * 16;
    const int lane = threadIdx.x & 31;
    const int r    = lane & 15;
    const int khalf = (lane >> 4) << 1;                // 0 or 2
    const int mhi   = (lane >> 4) << 3;                // 0 or 8

    const float* __restrict__ xrow = x + (size_t)(m0 + r) * DFEAT;
    const float* __restrict__ wrow = W + (size_t)(n0 + r) * DFEAT;

    v8f c = {};
#pragma unroll
    for (int k = 0; k < DFEAT; k += 4) {
        v2f a = *(const v2f*)(xrow + k + khalf);
        v2f b = *(const v2f*)(wrow + k + khalf);
        c = __builtin_amdgcn_wmma_f32_16x16x4_f32(
                /*neg_a=*/false, a, /*neg_b=*/false, b,
                /*c_mod=*/(short)0, c, /*reuse_a=*/false, /*reuse_b=*/false);
    }

    const float bv = bias[n0 + r];
#pragma unroll
    for (int v = 0; v < 8; ++v) {
        const int row = m0 + v + mhi;
        h[(size_t)row * DFEAT + n0 + r] = c[v] + bv;
    }
}

// Scalar fallback for a row remainder (not used when N % 16 == 0).
__global__ void linear_tail(const float* __restrict__ x,
                            const float* __restrict__ W,
                            const float* __restrict__ bias,
                            float* __restrict__ h,
                            int row_start, int N) {
    int i = blockIdx.x * blockDim.x + threadIdx.x;
    int row = row_start + i / DFEAT;
    int col = i % DFEAT;
    if (row >= N) return;
    float acc = bias[col];
    const float* xr = x + (size_t)row * DFEAT;
    const float* wr = W + (size_t)col * DFEAT;
    for (int k = 0; k < DFEAT; ++k) acc += xr[k] * wr[k];
    h[(size_t)row * DFEAT + col] = acc;
}

// ----------------------------------------------------------- degrees/counts
__global__ void degree_kernel(const int* __restrict__ src,
                              const int* __restrict__ dst,
                              float* __restrict__ deg,
                              float* __restrict__ cnt, int E) {
    int e = blockIdx.x * blockDim.x + threadIdx.x;
    if (e < E) {
        atomicAdd(&deg[src[e]], 1.0f);
        atomicAdd(&cnt[dst[e]], 1.0f);
    }
}

__global__ void rsqrt_kernel(const float* __restrict__ deg,
                             float* __restrict__ dis, int N) {
    int i = blockIdx.x * blockDim.x + threadIdx.x;
    if (i < N) dis[i] = rsqrtf(deg[i]);   // deg >= 1 always (self loops)
}

// -------------------------------------------------- edge scatter (messages)
// One wave per edge; lane handles features {2*lane, 2*lane+1}:
// a wave gathers one full 256B row of h (coalesced), scales by norm,
// scatter-adds into agg[dst] with global_atomic_add_f32.
__global__ void scatter_kernel(const int* __restrict__ src,
                               const int* __restrict__ dst,
                               const float* __restrict__ h,
                               const float* __restrict__ dis,
                               float* __restrict__ agg, int E) {
    long long i = (long long)blockIdx.x * blockDim.x + threadIdx.x;
    int e    = (int)(i >> 5);
    int lane = (int)(i & 31);
    if (e >= E) return;
    const int s = src[e];
    const int d = dst[e];
    const float norm = dis[s] * dis[d];
    const v2f hv = *(const v2f*)(h + (size_t)s * DFEAT + 2 * lane);
    float* ap = agg + (size_t)d * DFEAT + 2 * lane;
    atomicAdd(ap + 0, norm * hv.x);
    atomicAdd(ap + 1, norm * hv.y);
}

// -------------------------------- finalize: add self-loop term, divide mean
__global__ void finalize_kernel(float* __restrict__ out,
                                const float* __restrict__ h,
                                const float* __restrict__ dis,
                                const float* __restrict__ cnt, long long total) {
    long long i = (long long)blockIdx.x * blockDim.x + threadIdx.x;
    if (i >= total) return;
    const int n = (int)(i / DFEAT);
    const float dn = dis[n];
    out[i] = (out[i] + dn * dn * h[i]) / cnt[n];   // cnt >= 1 always
}

// --------------------------------------------------------------------------
extern "C" void kernel_launch(void* const* d_in, const int* in_sizes, int n_in,
                              void* d_out, int out_size, void* d_ws, size_t ws_size,
                              hipStream_t stream) {
    const float* x  = (const float*)d_in[0];
    const float* W  = (const float*)d_in[1];
    const float* b  = (const float*)d_in[2];
    const int*   ei = (const int*)d_in[3];

    const int N = in_sizes[0] / DFEAT;
    const int E = in_sizes[3] / 2;
    const int* src = ei;         // edge_index[0]
    const int* dst = ei + E;     // edge_index[1]

    float* out = (float*)d_out;
    float* h   = (float*)d_ws;                    // N*64
    float* deg = h + (size_t)N * DFEAT;           // N  (init 1.0: self loop)
    float* cnt = deg + N;                         // N  (init 1.0: self loop)
    float* dis = cnt + N;                         // N

    const long long nd = (long long)N * DFEAT;

    // init accumulators (d_out doubles as agg_sum buffer)
    fill_f32<<<(int)((nd + 255) / 256), 256, 0, stream>>>(out, 0.0f, (int)nd);
    fill_f32<<<(N + 255) / 256, 256, 0, stream>>>(deg, 1.0f, N);
    fill_f32<<<(N + 255) / 256, 256, 0, stream>>>(cnt, 1.0f, N);

    // h = x @ W^T + b   (WMMA, fp32)
    const int n_row_tiles = N / 16;
    if (n_row_tiles > 0) {
        dim3 grid((n_row_tiles + 3) / 4, DFEAT / 16);
        linear_wmma<<<grid, 128, 0, stream>>>(x, W, b, h, n_row_tiles);
    }
    const int rem = N % 16;
    if (rem > 0) {
        int tail_threads = rem * DFEAT;
        linear_tail<<<(tail_threads + 127) / 128, 128, 0, stream>>>(
            x, W, b, h, n_row_tiles * 16, N);
    }

    // degrees over src, counts over dst
    degree_kernel<<<(E + 255) / 256, 256, 0, stream>>>(src, dst, deg, cnt, E);
    rsqrt_kernel<<<(N + 255) / 256, 256, 0, stream>>>(deg, dis, N);

    // edge message scatter: one wave per edge
    const long long sthreads = (long long)E * 32;
    scatter_kernel<<<(int)((sthreads + 255) / 256), 256, 0, stream>>>(
        src, dst, h, dis, out, E);

    // self-loop term + mean
    finalize_kernel<<<(int)((nd + 255) / 256), 256, 0, stream>>>(out, h, dis, cnt, nd);
}